// LATTE_69965017252602
// MI455X (gfx1250) — compile-verified
//
#include <hip/hip_runtime.h>
#include <hip/hip_bf16.h>
#include <math.h>

// ---------------------------------------------------------------------------
// Problem constants (from reference)
// ---------------------------------------------------------------------------
static constexpr int N_   = 100000;
static constexpr int DIN_ = 512;
static constexpr int D_   = 512;
static constexpr int H_   = 8;
static constexpr int C_   = 64;   // D/H
static constexpr int M_   = 4;
static constexpr int R_   = 5;    // M+1
static constexpr int E_   = 150000;
static constexpr float NEG_ = 0.2f;

typedef __attribute__((ext_vector_type(2))) float v2f;
typedef __attribute__((ext_vector_type(8))) float v8f;

__device__ __forceinline__ float leaky(float v) {
    return v >= 0.0f ? v : NEG_ * v;
}

// atomic max for float via signed/unsigned int trick (slot must be inited to -inf)
__device__ __forceinline__ void atomicMaxFloat(float* addr, float value) {
    if (value >= 0.0f) {
        atomicMax((int*)addr, __float_as_int(value));
    } else {
        atomicMin((unsigned int*)addr, __float_as_uint(value));
    }
}

// ---------------------------------------------------------------------------
// Kernel 1: h = x @ W + b   (fp32 WMMA 16x16x4)
// grid: (N/16, 1), block: 256 (8 waves). Wave w computes col tiles w, w+8, w+16, w+24.
// A fragment (16x4 rows of x) loaded once per k-step and reused for 4 WMMAs.
// ---------------------------------------------------------------------------
__global__ void __launch_bounds__(256)
gemm_wmma_f32(const float* __restrict__ x, const float* __restrict__ W,
              const float* __restrict__ bias, float* __restrict__ h)
{
    const int wave = threadIdx.x >> 5;
    const int lane = threadIdx.x & 31;
    const int half = lane >> 4;    // 0: K pair {0,1} / rows 0..7 of D, 1: K pair {2,3} / rows 8..15
    const int l    = lane & 15;    // M index (A) / N index (B,C,D)

    const int rowBase = blockIdx.x * 16;
    const float* xr = x + (size_t)(rowBase + l) * DIN_;

    v8f acc0 = {}; v8f acc1 = {}; v8f acc2 = {}; v8f acc3 = {};

    const int col0 = (wave +  0) * 16 + l;
    const int col1 = (wave +  8) * 16 + l;
    const int col2 = (wave + 16) * 16 + l;
    const int col3 = (wave + 24) * 16 + l;

    for (int k = 0; k < DIN_; k += 4) {
        v2f a;
        a.x = xr[k + 2 * half + 0];
        a.y = xr[k + 2 * half + 1];

        const float* w0 = W + (size_t)(k + 2 * half + 0) * D_;
        const float* w1 = W + (size_t)(k + 2 * half + 1) * D_;

        v2f b0; b0.x = w0[col0]; b0.y = w1[col0];
        v2f b1; b1.x = w0[col1]; b1.y = w1[col1];
        v2f b2; b2.x = w0[col2]; b2.y = w1[col2];
        v2f b3; b3.x = w0[col3]; b3.y = w1[col3];

        acc0 = __builtin_amdgcn_wmma_f32_16x16x4_f32(false, a, false, b0, (short)0, acc0, false, false);
        acc1 = __builtin_amdgcn_wmma_f32_16x16x4_f32(false, a, false, b1, (short)0, acc1, false, false);
        acc2 = __builtin_amdgcn_wmma_f32_16x16x4_f32(false, a, false, b2, (short)0, acc2, false, false);
        acc3 = __builtin_amdgcn_wmma_f32_16x16x4_f32(false, a, false, b3, (short)0, acc3, false, false);
    }

    // C/D layout: lanes 0-15 -> M = vgpr (0..7); lanes 16-31 -> M = 8 + vgpr
    const float bv0 = bias[col0];
    const float bv1 = bias[col1];
    const float bv2 = bias[col2];
    const float bv3 = bias[col3];
#pragma unroll
    for (int r = 0; r < 8; ++r) {
        const size_t rowOff = (size_t)(rowBase + half * 8 + r) * D_;
        h[rowOff + col0] = acc0[r] + bv0;
        h[rowOff + col1] = acc1[r] + bv1;
        h[rowOff + col2] = acc2[r] + bv2;
        h[rowOff + col3] = acc3[r] + bv3;
    }
}

// ---------------------------------------------------------------------------
// Kernel 2: init scratch (mx = -inf, den = 0, rel_out = 0)
// ---------------------------------------------------------------------------
__global__ void init_scratch(float* __restrict__ mx, float* __restrict__ den,
                             float* __restrict__ rel_out)
{
    const size_t totalRel = (size_t)M_ * N_ * D_;
    const size_t totalMNH = (size_t)M_ * N_ * H_;
    const size_t stride = (size_t)gridDim.x * blockDim.x;
    for (size_t i = (size_t)blockIdx.x * blockDim.x + threadIdx.x; i < totalRel; i += stride) {
        rel_out[i] = 0.0f;
        if (i < totalMNH) { mx[i] = -INFINITY; den[i] = 0.0f; }
    }
}

// ---------------------------------------------------------------------------
// Kernel 3: per-node attention dot products
// s_l[m][n][hd] = sum_c h[n][hd][c] * attn[m][hd][c]
// s_r[m][n][hd] = sum_c h[n][hd][c] * attn[m][hd][C + c]
// one thread per (n, hd)
// ---------------------------------------------------------------------------
__global__ void node_dots(const float* __restrict__ h, const float* __restrict__ attn,
                          float* __restrict__ s_l, float* __restrict__ s_r)
{
    const int t = blockIdx.x * blockDim.x + threadIdx.x;
    if (t >= N_ * H_) return;
    const int n  = t / H_;
    const int hd = t % H_;

    float aL[M_] = {0.f, 0.f, 0.f, 0.f};
    float aR[M_] = {0.f, 0.f, 0.f, 0.f};
    const float* hp = h + (size_t)n * D_ + hd * C_;
    for (int c = 0; c < C_; ++c) {
        const float hv = hp[c];
#pragma unroll
        for (int m = 0; m < M_; ++m) {
            const float* ap = attn + ((size_t)m * H_ + hd) * (2 * C_);
            aL[m] += hv * ap[c];
            aR[m] += hv * ap[C_ + c];
        }
    }
#pragma unroll
    for (int m = 0; m < M_; ++m) {
        s_l[((size_t)m * N_ + n) * H_ + hd] = aL[m];
        s_r[((size_t)m * N_ + n) * H_ + hd] = aR[m];
    }
}

// ---------------------------------------------------------------------------
// Kernel 4: edge scores + segment max. one thread per (m, e)
// ---------------------------------------------------------------------------
__global__ void edge_score(const float* __restrict__ s_l, const float* __restrict__ s_r,
                           const int* __restrict__ esrc, const int* __restrict__ edst,
                           float* __restrict__ score, float* __restrict__ mx)
{
    const int t = blockIdx.x * blockDim.x + threadIdx.x;
    if (t >= M_ * E_) return;
    const int m   = t / E_;
    const int src = esrc[t];
    const int dst = edst[t];
    const float* slp = s_l + ((size_t)m * N_ + src) * H_;
    const float* srp = s_r + ((size_t)m * N_ + dst) * H_;
    float* mxp = mx + ((size_t)m * N_ + dst) * H_;
#pragma unroll
    for (int hd = 0; hd < H_; ++hd) {
        const float sc = leaky(slp[hd] + srp[hd]);
        score[(size_t)t * H_ + hd] = sc;
        atomicMaxFloat(&mxp[hd], sc);
    }
}

// ---------------------------------------------------------------------------
// Kernel 5: ex = exp(score - mx[dst]); den[dst] += ex. one thread per (m, e)
// score buffer is overwritten in-place with ex.
// ---------------------------------------------------------------------------
__global__ void edge_exp(const int* __restrict__ edst, const float* __restrict__ mx,
                         float* __restrict__ score, float* __restrict__ den)
{
    const int t = blockIdx.x * blockDim.x + threadIdx.x;
    if (t >= M_ * E_) return;
    const int m   = t / E_;
    const int dst = edst[t];
    const float* mxp = mx + ((size_t)m * N_ + dst) * H_;
    float* denp = den + ((size_t)m * N_ + dst) * H_;
#pragma unroll
    for (int hd = 0; hd < H_; ++hd) {
        const float ex = __expf(score[(size_t)t * H_ + hd] - mxp[hd]);
        score[(size_t)t * H_ + hd] = ex;
        atomicAdd(&denp[hd], ex);
    }
}

// ---------------------------------------------------------------------------
// Kernel 6: aggregation rel_out[m][dst] += h[src] * alpha
// 64 threads per edge (4 edges per 256-thread block). thread t covers
// idx = i*64 + t  (head = i, channel = t) for i in 0..7.
// ---------------------------------------------------------------------------
__global__ void __launch_bounds__(256)
edge_aggregate(const float* __restrict__ h, const float* __restrict__ ex,
               const float* __restrict__ den,
               const int* __restrict__ esrc, const int* __restrict__ edst,
               float* __restrict__ rel_out)
{
    const int eg = blockIdx.x * 4 + (threadIdx.x >> 6);
    if (eg >= M_ * E_) return;
    const int t   = threadIdx.x & 63;
    const int m   = eg / E_;
    const int src = esrc[eg];
    const int dst = edst[eg];

    const float* hp   = h + (size_t)src * D_;
    const float* denp = den + ((size_t)m * N_ + dst) * H_;
    float* outp = rel_out + ((size_t)m * N_ + dst) * D_;
#pragma unroll
    for (int i = 0; i < H_; ++i) {
        const float alpha = ex[(size_t)eg * H_ + i] / (denp[i] + 1e-16f);
        const int idx = i * 64 + t;
        atomicAdd(&outp[idx], hp[idx] * alpha);
    }
}

// ---------------------------------------------------------------------------
// Kernel 7: semantic attention + output. one 64-thread block per node.
// ---------------------------------------------------------------------------
__global__ void __launch_bounds__(64)
semantic_out(const float* __restrict__ h, const float* __restrict__ rel_out,
             const float* __restrict__ rel_attn_l, const float* __restrict__ rel_attn_r,
             const float* __restrict__ rel_attn_w, const float* __restrict__ rel_bias,
             float* __restrict__ out)
{
    const int n = blockIdx.x;
    const int t = threadIdx.x;          // 0..63
    const int head = t >> 3;            // 0..7
    const int sub  = t & 7;

    __shared__ float beta_s[R_ * H_];

    const float* hn = h + (size_t)n * D_;

    for (int r = 0; r < R_; ++r) {
        const float* hs = (r < M_) ? (rel_out + ((size_t)r * N_ + n) * D_) : hn;
        const float* al = rel_attn_l + ((size_t)r * H_ + head) * C_;
        const float* ar = rel_attn_r + ((size_t)r * H_ + head) * C_;
        float p = 0.0f;
#pragma unroll
        for (int j = 0; j < 8; ++j) {
            const int c = sub + 8 * j;
            const int idx = head * C_ + c;
            p += hn[idx] * al[c] + hs[idx] * ar[c];
        }
        // reduce within contiguous groups of 8 lanes (wave32)
        p += __shfl_xor(p, 4, 32);
        p += __shfl_xor(p, 2, 32);
        p += __shfl_xor(p, 1, 32);
        if (sub == 0) beta_s[r * H_ + head] = leaky(p) + rel_bias[r];
    }
    __syncthreads();

    // softmax across relations, per head (threads 0..7)
    if (t < H_) {
        float m = -INFINITY;
#pragma unroll
        for (int r = 0; r < R_; ++r) m = fmaxf(m, beta_s[r * H_ + t]);
        float s = 0.0f;
        float e[R_];
#pragma unroll
        for (int r = 0; r < R_; ++r) { e[r] = __expf(beta_s[r * H_ + t] - m); s += e[r]; }
        const float inv = 1.0f / s;
#pragma unroll
        for (int r = 0; r < R_; ++r) beta_s[r * H_ + t] = e[r] * inv;
    }
    __syncthreads();

    // out[n][idx] = relu( sum_r hs[r][idx] * w[r][idx] * beta[r][head(idx)] )
#pragma unroll
    for (int i = 0; i < H_; ++i) {
        const int idx = i * 64 + t;      // head = i, channel = t
        float acc = 0.0f;
#pragma unroll
        for (int r = 0; r < R_; ++r) {
            const float* hs = (r < M_) ? (rel_out + ((size_t)r * N_ + n) * D_) : hn;
            const float w = (r < M_) ? rel_attn_w[((size_t)r * H_ + i) * C_ + t] : 1.0f;
            acc += hs[idx] * w * beta_s[r * H_ + i];
        }
        out[(size_t)n * D_ + idx] = fmaxf(acc, 0.0f);
    }
}

// ---------------------------------------------------------------------------
// Launch
// ---------------------------------------------------------------------------
extern "C" void kernel_launch(void* const* d_in, const int* in_sizes, int n_in,
                              void* d_out, int out_size, void* d_ws, size_t ws_size,
                              hipStream_t stream) {
    const float* x          = (const float*)d_in[0];
    const float* W          = (const float*)d_in[1];
    const float* bias       = (const float*)d_in[2];
    const float* attn       = (const float*)d_in[3];
    const float* rel_attn_l = (const float*)d_in[4];
    const float* rel_attn_r = (const float*)d_in[5];
    const float* rel_attn_w = (const float*)d_in[6];
    const float* rel_bias   = (const float*)d_in[7];
    const int*   edge_src   = (const int*)d_in[8];
    const int*   edge_dst   = (const int*)d_in[9];
    float* out = (float*)d_out;

    // workspace layout (floats)
    float* ws    = (float*)d_ws;
    float* h     = ws;                                   // N*D
    float* s_l   = h     + (size_t)N_ * D_;              // M*N*H
    float* s_r   = s_l   + (size_t)M_ * N_ * H_;         // M*N*H
    float* mx    = s_r   + (size_t)M_ * N_ * H_;         // M*N*H
    float* den   = mx    + (size_t)M_ * N_ * H_;         // M*N*H
    float* score = den   + (size_t)M_ * N_ * H_;         // M*E*H (score -> ex)
    float* rel   = score + (size_t)M_ * E_ * H_;         // M*N*D

    // 1) GEMM h = x@W + b (WMMA f32)
    gemm_wmma_f32<<<dim3(N_ / 16, 1), dim3(256), 0, stream>>>(x, W, bias, h);

    // 2) init mx/den/rel_out
    init_scratch<<<dim3(4096), dim3(256), 0, stream>>>(mx, den, rel);

    // 3) per-node attention dot products
    node_dots<<<dim3((N_ * H_ + 255) / 256), dim3(256), 0, stream>>>(h, attn, s_l, s_r);

    // 4) edge scores + segment max
    edge_score<<<dim3((M_ * E_ + 255) / 256), dim3(256), 0, stream>>>(
        s_l, s_r, edge_src, edge_dst, score, mx);

    // 5) exp + segment sum
    edge_exp<<<dim3((M_ * E_ + 255) / 256), dim3(256), 0, stream>>>(
        edge_dst, mx, score, den);

    // 6) weighted aggregation into rel_out
    edge_aggregate<<<dim3((M_ * E_ + 3) / 4), dim3(256), 0, stream>>>(
        h, score, den, edge_src, edge_dst, rel);

    // 7) semantic attention + relu output
    semantic_out<<<dim3(N_), dim3(64), 0, stream>>>(
        h, rel, rel_attn_l, rel_attn_r, rel_attn_w, rel_bias, out);
}